// NeuralFingerprint_63874753626408
// MI455X (gfx1250) — compile-verified
//
#include <hip/hip_runtime.h>
#include <hip/hip_bf16.h>

#define NODE 128
#define EDGEF 16
#define OUTF 512
#define NN 131072
#define EE 524288
#define BB 4096
#define DD 6
#define LL 3
#define EPSV 1e-5f
#define NBRF 144   // NODE + EDGEF

typedef __attribute__((ext_vector_type(2))) float v2f;
typedef __attribute__((ext_vector_type(8))) float v8f;

__device__ __forceinline__ v8f wmma_f32(v2f a, v2f b, v8f c) {
  // D = A(16x4) * B(4x16) + C, fp32
  return __builtin_amdgcn_wmma_f32_16x16x4_f32(false, a, false, b, (short)0, c,
                                               false, false);
}

// ---- async global->LDS copy (CDNA5, ASYNCcnt-tracked, bypasses VGPRs) ----
__device__ __forceinline__ uint32_t lds_addr32(const void* p) {
  // flat shared pointer: low 32 bits are the LDS offset (aperture rule)
  return (uint32_t)(uintptr_t)p;
}
__device__ __forceinline__ void async_copy_b128(uint32_t lds_off, const void* g) {
  asm volatile("global_load_async_to_lds_b128 %0, %1, off"
               :
               : "v"(lds_off), "v"(g)
               : "memory");
}
__device__ __forceinline__ void async_wait0() {
  asm volatile("s_wait_asynccnt 0x0" ::: "memory");
}

// ---------------- utility kernels ----------------
__global__ void k_fill_f(float* __restrict__ p, int n, float v) {
  int i = blockIdx.x * 256 + threadIdx.x;
  if (i < n) p[i] = v;
}
__global__ void k_fill_i(int* __restrict__ p, int n, int v) {
  int i = blockIdx.x * 256 + threadIdx.x;
  if (i < n) p[i] = v;
}

// ---------------- degree + bucketing (once per call) ----------------
__global__ void k_count_deg(const int* __restrict__ dst, int* __restrict__ deg) {
  int e = blockIdx.x * 256 + threadIdx.x;
  if (e < EE) atomicAdd(&deg[dst[e]], 1);
}
__global__ void k_bucket_count(const int* __restrict__ deg, int* __restrict__ cnt) {
  int n = blockIdx.x * 256 + threadIdx.x;
  if (n < NN) atomicAdd(&cnt[min(deg[n], DD - 1)], 1);
}
__global__ void k_offsets(const int* __restrict__ cnt, int* __restrict__ off) {
  if (threadIdx.x == 0 && blockIdx.x == 0) {
    int o = 0;
    off[0] = 0;
    for (int d = 0; d < DD; d++) {
      o += ((cnt[d] + 15) >> 4) << 4;  // pad each bucket to 16-row tiles
      off[d + 1] = o;
    }
  }
}
__global__ void k_perm_scatter(const int* __restrict__ deg, const int* __restrict__ off,
                               int* __restrict__ cursor, int* __restrict__ perm) {
  int n = blockIdx.x * 256 + threadIdx.x;
  if (n < NN) {
    int d = min(deg[n], DD - 1);
    int p = off[d] + atomicAdd(&cursor[d], 1);
    perm[p] = n;
  }
}

// ---------------- CSR build: exclusive scan of deg -> edge_off ----------------
__global__ __launch_bounds__(256) void k_scan1(const int* __restrict__ deg,
                                               int* __restrict__ eoff,
                                               int* __restrict__ bsum) {
  __shared__ int sh[256];
  int i = blockIdx.x * 256 + threadIdx.x;
  int v = deg[i];
  sh[threadIdx.x] = v;
  __syncthreads();
  for (int ofs = 1; ofs < 256; ofs <<= 1) {
    int x = (threadIdx.x >= ofs) ? sh[threadIdx.x - ofs] : 0;
    __syncthreads();
    sh[threadIdx.x] += x;
    __syncthreads();
  }
  eoff[i] = sh[threadIdx.x] - v;  // exclusive
  if (threadIdx.x == 255) bsum[blockIdx.x] = sh[255];
}
__global__ __launch_bounds__(512) void k_scan2(const int* __restrict__ bsum,
                                               int* __restrict__ bpre) {
  __shared__ int sh[512];
  int t = threadIdx.x;
  int v = bsum[t];
  sh[t] = v;
  __syncthreads();
  for (int ofs = 1; ofs < 512; ofs <<= 1) {
    int x = (t >= ofs) ? sh[t - ofs] : 0;
    __syncthreads();
    sh[t] += x;
    __syncthreads();
  }
  bpre[t] = sh[t] - v;  // exclusive
}
__global__ __launch_bounds__(256) void k_scan3(int* __restrict__ eoff,
                                               const int* __restrict__ bpre) {
  int i = blockIdx.x * 256 + threadIdx.x;
  eoff[i] += bpre[blockIdx.x];
}
__global__ void k_edge_sort(const int* __restrict__ dst, const int* __restrict__ eoff,
                            int* __restrict__ cur, int* __restrict__ eperm) {
  int e = blockIdx.x * 256 + threadIdx.x;
  if (e < EE) {
    int d = dst[e];
    int pos = eoff[d] + atomicAdd(&cur[d], 1);
    eperm[pos] = e;
  }
}

// ---------------- fingerprint update: softmax(h @ W + b) segment-summed ----------------
__global__ __launch_bounds__(256) void k_fp(const float* __restrict__ h,
                                            const float* __restrict__ W,
                                            const float* __restrict__ bias,
                                            const int* __restrict__ gids,
                                            float* __restrict__ out) {
  __shared__ alignas(16) float As[16][132];   // 16 x 128 A tile (padded)
  __shared__ alignas(16) float Ps[16][520];   // 16 x 512 result tile (padded)
  __shared__ int gid_s[16];
  const int tid = threadIdx.x;
  const int wave = tid >> 5, lane = tid & 31;
  const int half = lane >> 4, lh = lane & 15;
  const int m0 = blockIdx.x * 16;

  // async-copy A tile: 16 rows x 32 b128 chunks
  for (int i = tid; i < 16 * 32; i += 256) {
    int r = i >> 5, c = i & 31;
    async_copy_b128(lds_addr32(&As[r][c * 4]),
                    h + (size_t)(m0 + r) * NODE + c * 4);
  }
  if (tid < 16) gid_s[tid] = gids[m0 + tid];
  async_wait0();
  __syncthreads();

  v8f acc[4];
#pragma unroll
  for (int j = 0; j < 4; j++)
#pragma unroll
    for (int i = 0; i < 8; i++) acc[j][i] = 0.f;

  // each wave: 4 column tiles n0 = wave*64 + j*16, K = 128
  for (int kb = 0; kb < NODE; kb += 4) {
    v2f a;
    a.x = As[lh][kb + 2 * half];
    a.y = As[lh][kb + 2 * half + 1];
#pragma unroll
    for (int j = 0; j < 4; j++) {
      int n0 = wave * 64 + j * 16;
      v2f b;
      b.x = W[(size_t)(kb + 2 * half) * OUTF + n0 + lh];
      b.y = W[(size_t)(kb + 2 * half + 1) * OUTF + n0 + lh];
      acc[j] = wmma_f32(a, b, acc[j]);
    }
  }

  // spill to LDS with bias (C layout: VGPR r -> rows r / r+8)
#pragma unroll
  for (int j = 0; j < 4; j++) {
    int n0 = wave * 64 + j * 16;
    float bv = bias[n0 + lh];
#pragma unroll
    for (int r = 0; r < 8; r++) Ps[r + 8 * half][n0 + lh] = acc[j][r] + bv;
  }
  __syncthreads();

  // stable softmax: wave w handles rows w and w+8
  for (int rr = 0; rr < 2; rr++) {
    int r = wave + rr * 8;
    float mx = -1e30f;
    for (int c = lane; c < OUTF; c += 32) mx = fmaxf(mx, Ps[r][c]);
    for (int s = 16; s > 0; s >>= 1) mx = fmaxf(mx, __shfl_xor(mx, s, 32));
    float sum = 0.f;
    for (int c = lane; c < OUTF; c += 32) {
      float v = expf(Ps[r][c] - mx);
      Ps[r][c] = v;
      sum += v;
    }
    for (int s = 16; s > 0; s >>= 1) sum += __shfl_xor(sum, s, 32);
    float inv = 1.f / sum;
    for (int c = lane; c < OUTF; c += 32) Ps[r][c] *= inv;
  }
  __syncthreads();

  // segment-sum over sorted graph ids: run-length accumulate, few atomics
  for (int c = tid; c < OUTF; c += 256) {
    int cur = gid_s[0];
    float a = 0.f;
    for (int r = 0; r < 16; r++) {
      int g = gid_s[r];
      if (g != cur) {
        atomicAdd(&out[(size_t)cur * OUTF + c], a);
        a = 0.f;
        cur = g;
      }
      a += Ps[r][c];
    }
    atomicAdd(&out[(size_t)cur * OUTF + c], a);
  }
}

// ---------------- neighbor sum via CSR: one wave per node, no atomics ----------------
__global__ __launch_bounds__(256) void k_nbr(const float* __restrict__ h,
                                             const float* __restrict__ bond,
                                             const int* __restrict__ esrc,
                                             const int* __restrict__ eperm,
                                             const int* __restrict__ eoff,
                                             const int* __restrict__ deg,
                                             float* __restrict__ nbr) {
  int n = blockIdx.x * 8 + (threadIdx.x >> 5);
  int lane = threadIdx.x & 31;
  int beg = eoff[n];
  int end = beg + deg[n];
  float a0 = 0.f, a1 = 0.f, a2 = 0.f, a3 = 0.f, ab = 0.f;
  for (int j = beg; j < end; j++) {
    int e = eperm[j];
    int s = esrc[e];
    const float* hs = h + (size_t)s * NODE;
    a0 += hs[lane];
    a1 += hs[32 + lane];
    a2 += hs[64 + lane];
    a3 += hs[96 + lane];
    if (lane < EDGEF) ab += bond[(size_t)e * EDGEF + lane];
  }
  float* nb = nbr + (size_t)n * NBRF;
  nb[lane] = a0;
  nb[32 + lane] = a1;
  nb[64 + lane] = a2;
  nb[96 + lane] = a3;
  if (lane < EDGEF) nb[NODE + lane] = ab;
}

// ---------------- conv GEMM: act[n] = h[n]@sW + nbr_sum[n]@dW[deg[n]] (bucketed tiles) ----------------
__global__ __launch_bounds__(256) void k_conv(const float* __restrict__ h,
                                              const float* __restrict__ nbr,
                                              const float* __restrict__ sW,
                                              const float* __restrict__ dW,
                                              const int* __restrict__ perm,
                                              const int* __restrict__ off,
                                              float* __restrict__ act) {
  __shared__ alignas(16) float Ah[16][132];  // h rows
  __shared__ alignas(16) float An[16][148];  // nbr_sum rows
  __shared__ int idx_s[16];
  const int t = blockIdx.x;
  const int row0 = t * 16;
  if (row0 >= off[DD]) return;  // block-uniform: past last real bucket
  int d = 0;
  while (d < DD - 1 && row0 >= off[d + 1]) d++;

  const int tid = threadIdx.x;
  const int wave = tid >> 5, lane = tid & 31;
  const int half = lane >> 4, lh = lane & 15;
  if (tid < 16) idx_s[tid] = perm[row0 + tid];
  __syncthreads();

  {
    int r = tid >> 4;    // 16 threads per row
    int c0 = tid & 15;
    int node = idx_s[r];
    if (node >= 0) {
      const float* hr = h + (size_t)node * NODE;
      const float* nr = nbr + (size_t)node * NBRF;
      for (int c = c0; c < 32; c += 16)
        async_copy_b128(lds_addr32(&Ah[r][c * 4]), hr + c * 4);
      for (int c = c0; c < 36; c += 16)
        async_copy_b128(lds_addr32(&An[r][c * 4]), nr + c * 4);
    } else {
      for (int c = c0; c < NODE; c += 16) Ah[r][c] = 0.f;
      for (int c = c0; c < NBRF; c += 16) An[r][c] = 0.f;
    }
  }
  async_wait0();
  __syncthreads();

  const int n0 = wave * 16;  // 8 waves cover 128 output cols
  v8f acc;
#pragma unroll
  for (int i = 0; i < 8; i++) acc[i] = 0.f;

  for (int kb = 0; kb < NODE; kb += 4) {
    v2f a, b;
    a.x = Ah[lh][kb + 2 * half];
    a.y = Ah[lh][kb + 2 * half + 1];
    b.x = sW[(size_t)(kb + 2 * half) * NODE + n0 + lh];
    b.y = sW[(size_t)(kb + 2 * half + 1) * NODE + n0 + lh];
    acc = wmma_f32(a, b, acc);
  }
  const float* Wd = dW + (size_t)d * NBRF * NODE;
  for (int kb = 0; kb < NBRF; kb += 4) {
    v2f a, b;
    a.x = An[lh][kb + 2 * half];
    a.y = An[lh][kb + 2 * half + 1];
    b.x = Wd[(size_t)(kb + 2 * half) * NODE + n0 + lh];
    b.y = Wd[(size_t)(kb + 2 * half + 1) * NODE + n0 + lh];
    acc = wmma_f32(a, b, acc);
  }

#pragma unroll
  for (int r = 0; r < 8; r++) {
    int row = r + 8 * half;
    int node = idx_s[row];
    if (node >= 0) act[(size_t)node * NODE + n0 + lh] = acc[r];
  }
}

// ---------------- batch-norm statistics (sum / sumsq per feature) ----------------
__global__ __launch_bounds__(256) void k_bnstats(const float* __restrict__ act,
                                                 float* __restrict__ stats) {
  const int f = threadIdx.x & 127;
  const int sub = threadIdx.x >> 7;            // 0/1
  const int rows = NN / 512;                   // grid = 512 blocks
  const int r0 = blockIdx.x * rows;
  float s = 0.f, q = 0.f;
  for (int r = r0 + sub; r < r0 + rows; r += 2) {
    float v = act[(size_t)r * NODE + f];
    s += v;
    q += v * v;
  }
  __shared__ float shs[256], shq[256];
  shs[threadIdx.x] = s;
  shq[threadIdx.x] = q;
  __syncthreads();
  if (sub == 0) {
    s += shs[threadIdx.x + 128];
    q += shq[threadIdx.x + 128];
    atomicAdd(&stats[f], s);
    atomicAdd(&stats[128 + f], q);
  }
}

// ---------------- normalize + relu (in place) ----------------
__global__ void k_bnapply(float* __restrict__ act, const float* __restrict__ stats) {
  size_t i = (size_t)blockIdx.x * 256 + threadIdx.x;
  if (i >= (size_t)NN * NODE) return;
  int f = (int)(i & 127);
  float mean = stats[f] * (1.f / NN);
  float var = stats[128 + f] * (1.f / NN) - mean * mean;
  float v = (act[i] - mean) * rsqrtf(var + EPSV);
  act[i] = fmaxf(v, 0.f);
}

// ---------------- host orchestration ----------------
extern "C" void kernel_launch(void* const* d_in, const int* in_sizes, int n_in,
                              void* d_out, int out_size, void* d_ws, size_t ws_size,
                              hipStream_t stream) {
  const float* atom = (const float*)d_in[0];
  const float* bond = (const float*)d_in[1];
  const float* out_W = (const float*)d_in[2];   // [4,128,512]
  const float* out_b = (const float*)d_in[3];   // [4,512]
  const float* self_W = (const float*)d_in[4];  // [3,128,128]
  const float* deg_W = (const float*)d_in[5];   // [3,6,144,128]
  const int* edge_src = (const int*)d_in[6];
  const int* edge_dst = (const int*)d_in[7];
  const int* graph_ids = (const int*)d_in[8];
  float* out = (float*)d_out;

  char* ws = (char*)d_ws;
  float* bufA = (float*)ws;  ws += (size_t)NN * NODE * 4;
  float* bufB = (float*)ws;  ws += (size_t)NN * NODE * 4;
  float* nbr = (float*)ws;   ws += (size_t)NN * NBRF * 4;
  int* deg = (int*)ws;       ws += (size_t)NN * 4;
  int* perm = (int*)ws;      ws += (size_t)(NN + 16 * DD) * 4;
  int* eoff = (int*)ws;      ws += (size_t)NN * 4;
  int* cur_n = (int*)ws;     ws += (size_t)NN * 4;
  int* eperm = (int*)ws;     ws += (size_t)EE * 4;
  int* bsum = (int*)ws;      ws += 512 * 4;
  int* bpre = (int*)ws;      ws += 512 * 4;
  int* cnt = (int*)ws;       ws += 16 * 4;
  int* off = (int*)ws;       ws += 16 * 4;
  int* cursor = (int*)ws;    ws += 16 * 4;
  float* stats = (float*)ws; ws += 256 * 4;

  const int PERM_N = NN + 16 * DD;

  // --- once-per-call setup: degrees, degree-bucketed permutation, edge CSR ---
  k_fill_i<<<(NN + 255) / 256, 256, 0, stream>>>(deg, NN, 0);
  k_fill_i<<<(NN + 255) / 256, 256, 0, stream>>>(cur_n, NN, 0);
  k_fill_i<<<1, 64, 0, stream>>>(cnt, 16, 0);
  k_fill_i<<<1, 64, 0, stream>>>(cursor, 16, 0);
  k_count_deg<<<(EE + 255) / 256, 256, 0, stream>>>(edge_dst, deg);
  k_bucket_count<<<NN / 256, 256, 0, stream>>>(deg, cnt);
  k_offsets<<<1, 1, 0, stream>>>(cnt, off);
  k_fill_i<<<(PERM_N + 255) / 256, 256, 0, stream>>>(perm, PERM_N, -1);
  k_perm_scatter<<<NN / 256, 256, 0, stream>>>(deg, off, cursor, perm);
  k_scan1<<<512, 256, 0, stream>>>(deg, eoff, bsum);
  k_scan2<<<1, 512, 0, stream>>>(bsum, bpre);
  k_scan3<<<512, 256, 0, stream>>>(eoff, bpre);
  k_edge_sort<<<EE / 256, 256, 0, stream>>>(edge_dst, eoff, cur_n, eperm);
  k_fill_f<<<(BB * OUTF + 255) / 256, 256, 0, stream>>>(out, BB * OUTF, 0.f);

  const float* h = atom;
  float* nxt = bufA;
  for (int l = 0; l < LL; l++) {
    k_fp<<<NN / 16, 256, 0, stream>>>(h, out_W + (size_t)l * NODE * OUTF,
                                      out_b + (size_t)l * OUTF, graph_ids, out);
    k_nbr<<<NN / 8, 256, 0, stream>>>(h, bond, edge_src, eperm, eoff, deg, nbr);
    k_fill_f<<<1, 256, 0, stream>>>(stats, 256, 0.f);
    k_conv<<<PERM_N / 16, 256, 0, stream>>>(
        h, nbr, self_W + (size_t)l * NODE * NODE,
        deg_W + (size_t)l * DD * NBRF * NODE, perm, off, nxt);
    k_bnstats<<<512, 256, 0, stream>>>(nxt, stats);
    k_bnapply<<<(int)(((size_t)NN * NODE + 255) / 256), 256, 0, stream>>>(nxt, stats);
    h = nxt;
    nxt = (l == 0) ? bufB : bufA;
  }
  k_fp<<<NN / 16, 256, 0, stream>>>(h, out_W + (size_t)LL * NODE * OUTF,
                                    out_b + (size_t)LL * OUTF, graph_ids, out);
}